// Criterion_67954972557710
// MI455X (gfx1250) — compile-verified
//
#include <hip/hip_runtime.h>

// CDNA5 / gfx1250: wave32, WMMA fragments.
typedef float v2f __attribute__((ext_vector_type(2)));
typedef float v8f __attribute__((ext_vector_type(8)));

#define THICKNESS 0.00047f
#define BLOCK 256   // 8 wave32 waves per workgroup

// One face per thread. Memory-bound: ~52B streamed + ~36B gathered per face.
// Scatter uses the identity mean_j( pv[:,j] ) == (1/3) * sum of all scatters
// hitting a vertex, so a single f32 atomic accumulator per vertex suffices.
__global__ void __launch_bounds__(BLOCK)
stvk_face_energy_kernel(const float* __restrict__ pred_pos,
                        const long long* __restrict__ faces,
                        const float* __restrict__ Dm_inv,
                        const float* __restrict__ f_area,
                        const float* __restrict__ lame_mu,
                        const float* __restrict__ lame_lambda,
                        float* __restrict__ loss,       // d_out[0]
                        float* __restrict__ per_vert,   // d_out + 1
                        int F)
{
    __shared__ float wsum[BLOCK / 32];

    const int f   = blockIdx.x * BLOCK + threadIdx.x;
    const float mu  = lame_mu[0];       // uniform -> scalar loads
    const float lam = lame_lambda[0];

    float energy = 0.0f;

    if (f < F) {
        // face indices: 3 x int64, 8B-aligned (b64 loads)
        const long long* fr = faces + 3ll * f;
        const int v0 = (int)fr[0];
        const int v1 = (int)fr[1];
        const int v2 = (int)fr[2];

        // Dm_inv row-major [2][2]: one global_load_b128
        const float4 dm = *(const float4*)(Dm_inv + 4ll * f);
        const float area = f_area[f];

        // gather 3 vertices (3 floats each); L2-resident (12MB array)
        const float* p0 = pred_pos + 3ll * (long long)v0;
        const float* p1 = pred_pos + 3ll * (long long)v1;
        const float* p2 = pred_pos + 3ll * (long long)v2;

        // Ds columns: a = v0 - v2, b = v1 - v2
        const float a0 = p0[0] - p2[0], a1 = p0[1] - p2[1], a2 = p0[2] - p2[2];
        const float b0 = p1[0] - p2[0], b1 = p1[1] - p2[1], b2 = p1[2] - p2[2];

        // Fg = Ds @ Dm_inv  (columns u, w of the 3x2 deformation gradient)
        // dm = [d00, d01, d10, d11]
        const float u0 = a0 * dm.x + b0 * dm.z;
        const float u1 = a1 * dm.x + b1 * dm.z;
        const float u2 = a2 * dm.x + b2 * dm.z;
        const float w0 = a0 * dm.y + b0 * dm.w;
        const float w1 = a1 * dm.y + b1 * dm.w;
        const float w2 = a2 * dm.y + b2 * dm.w;

        // Green strain G = 0.5 (F^T F - I)
        const float uu = u0 * u0 + u1 * u1 + u2 * u2;
        const float uw = u0 * w0 + u1 * w1 + u2 * w2;
        const float ww = w0 * w0 + w1 * w1 + w2 * w2;
        const float g00 = 0.5f * (uu - 1.0f);
        const float g01 = 0.5f * uw;
        const float g11 = 0.5f * (ww - 1.0f);
        const float tr  = g00 + g11;

        // StVK stress S = mu*G + 0.5*lam*tr(G)*I ; density = sum S_ij G_ij
        const float hlt = 0.5f * lam * tr;
        const float s00 = mu * g00 + hlt;
        const float s01 = mu * g01;
        const float s11 = mu * g11 + hlt;
        const float density = s00 * g00 + 2.0f * s01 * g01 + s11 * g11;

        energy = area * THICKNESS * density;

        // scatter: per_vert[v] accumulates energy/3 (mean over the 3 columns)
        const float e3 = energy * (1.0f / 3.0f);
        atomicAdd(per_vert + v0, e3);
        atomicAdd(per_vert + v1, e3);
        atomicAdd(per_vert + v2, e3);
    }

    // ---- wave-level loss reduction via V_WMMA_F32_16X16X4_F32 ----
    // A = ones(16x4). Each lane contributes its energy in one distinct B slot
    // (B VGPR1 = 0 zeroes the remaining K slots). Then
    //   C[m,n] = sum_k A[m,k]*B[k,n] = colsum(n)
    // and sum_n colsum(n) over the 16 columns == sum of all 32 lane energies,
    // regardless of which (k,n) slot each lane's VGPR0 maps to.
    // All BLOCK threads reach this point (no early return) => EXEC all-ones.
    {
        const v2f a_ones = {1.0f, 1.0f};
        v2f b_e;
        b_e[0] = energy;   // one distinct B slot per lane
        b_e[1] = 0.0f;     // remaining K slots zero
        v8f c = {};
        c = __builtin_amdgcn_wmma_f32_16x16x4_f32(
                /*neg_a=*/false, a_ones,
                /*neg_b=*/false, b_e,
                /*c_mod=*/(short)0, c,
                /*reuse_a=*/false, /*reuse_b=*/false);

        // c[0] in every lane = colsum(N = lane%16); fold 16 columns -> 1
        float s = c[0];
        s += __shfl_xor(s, 1, 32);
        s += __shfl_xor(s, 2, 32);
        s += __shfl_xor(s, 4, 32);
        s += __shfl_xor(s, 8, 32);

        const int lane = threadIdx.x & 31;
        const int wid  = threadIdx.x >> 5;
        if (lane == 0) wsum[wid] = s;
        __syncthreads();
        if (threadIdx.x == 0) {
            float bs = 0.0f;
            #pragma unroll
            for (int i = 0; i < BLOCK / 32; ++i) bs += wsum[i];
            atomicAdd(loss, bs);   // B == 1, so loss = sum(energy)
        }
    }
}

extern "C" void kernel_launch(void* const* d_in, const int* in_sizes, int n_in,
                              void* d_out, int out_size, void* d_ws, size_t ws_size,
                              hipStream_t stream) {
    (void)n_in; (void)d_ws; (void)ws_size;

    const float*     pred_pos = (const float*)d_in[0];
    const long long* faces    = (const long long*)d_in[1];   // int64 in reference
    const float*     Dm_inv   = (const float*)d_in[2];
    const float*     f_area   = (const float*)d_in[3];
    const float*     lame_mu  = (const float*)d_in[4];
    const float*     lame_lam = (const float*)d_in[5];

    const int F = in_sizes[3];          // f_area has F elements

    float* out      = (float*)d_out;    // [0] = loss, [1..V] = per_vert
    float* loss     = out;
    float* per_vert = out + 1;

    // Zero the atomic accumulators every call (capture-legal memset node).
    hipMemsetAsync(d_out, 0, (size_t)out_size * sizeof(float), stream);

    const int nblocks = (F + BLOCK - 1) / BLOCK;
    stvk_face_energy_kernel<<<nblocks, BLOCK, 0, stream>>>(
        pred_pos, faces, Dm_inv, f_area, lame_mu, lame_lam,
        loss, per_vert, F);
}